// GAT_25134148616643
// MI455X (gfx1250) — compile-verified
//
#include <hip/hip_runtime.h>
#include <math.h>

// ---------------------------------------------------------------------------
// GAT (3-layer) for MI455X / gfx1250.
//   N=50000 nodes, E=1.6M edges (+N self loops), F_IN=128.
//   L1: 128->4x16 concat, L2: 64->4x16 concat, L3: 64->6x40 mean, log_softmax.
// Dense GEMMs use V_WMMA_F32_16X16X4_F32 (precision-exact matrix path).
// Edge phase: segment softmax (atomic-max via ordered-uint encoding,
// atomic-add f32) with acc arrays sized to stay resident in the 192MB L2.
// ---------------------------------------------------------------------------

typedef __attribute__((ext_vector_type(2))) float v2f;
typedef __attribute__((ext_vector_type(8))) float v8f;

#define N_NODES   50000
#define N_EDGES   1600000
#define E_TOT     (N_EDGES + N_NODES)

// ---------------- ordered-uint encoding for float atomic max ----------------
__device__ __forceinline__ unsigned flipf(float f) {
  unsigned u = __float_as_uint(f);
  return (u & 0x80000000u) ? ~u : (u | 0x80000000u);
}
__device__ __forceinline__ float unflipu(unsigned u) {
  return __uint_as_float((u & 0x80000000u) ? (u ^ 0x80000000u) : ~u);
}
#define NEG_INF_KEY 0x007FFFFFu   // flipf(-inf)

__device__ __forceinline__ float leaky02(float x) { return x > 0.f ? x : 0.2f * x; }
__device__ __forceinline__ float elu1(float x)    { return x > 0.f ? x : expf(x) - 1.f; }

__device__ __forceinline__ void edge_sd(const int* __restrict__ ei, int e,
                                        int& s, int& d) {
  if (e < N_EDGES) { s = ei[e]; d = ei[N_EDGES + e]; }
  else             { s = d = e - N_EDGES; }
}

// ---------------------------------------------------------------------------
// GEMM: H[N, FOUT] = X[N, FIN] @ W[FIN, FOUT]   (one 16x16 tile per wave)
// A frag (16x4 f32): lane l<16 holds X[row0+l][k+0..1]; l>=16 -> K=k+2..3
// B frag (4x16 f32): VGPR j, lanes 0-15: K=k+j, N=lane; lanes16-31: K=k+2+j
// C/D (16x16 f32):   VGPR v, lanes 0-15: M=v, lanes 16-31: M=v+8, N=lane&15
// ---------------------------------------------------------------------------
template <int FIN, int FOUT>
__global__ __launch_bounds__(128) void gat_gemm_wmma(
    const float* __restrict__ X, const float* __restrict__ W,
    float* __restrict__ Hout, int nTiles) {
  const int wave = threadIdx.x >> 5;
  const int lane = threadIdx.x & 31;
  const int tile = blockIdx.x * 4 + wave;           // wave-uniform
  if (tile >= nTiles) return;                       // whole wave exits -> EXEC full
  const int tilesCol = FOUT / 16;
  const int row0 = (tile / tilesCol) * 16;
  const int col0 = (tile % tilesCol) * 16;
  const int half = lane >> 4;                       // 0 or 1
  const int l    = lane & 15;

#if __has_builtin(__builtin_amdgcn_wmma_f32_16x16x4_f32)
  v8f acc = {};
  const float* __restrict__ xrow = X + (long)(row0 + l) * FIN + 2 * half;
#pragma unroll
  for (int k = 0; k < FIN; k += 4) {
    v2f a, b;
    a.x = xrow[k + 0];
    a.y = xrow[k + 1];
    b.x = W[(long)(k + 2 * half + 0) * FOUT + col0 + l];
    b.y = W[(long)(k + 2 * half + 1) * FOUT + col0 + l];
    acc = __builtin_amdgcn_wmma_f32_16x16x4_f32(
        /*neg_a=*/false, a, /*neg_b=*/false, b,
        /*c_mod=*/(short)0, acc, /*reuse_a=*/false, /*reuse_b=*/false);
  }
#pragma unroll
  for (int v = 0; v < 8; ++v) {
    int r = row0 + v + half * 8;
    Hout[(long)r * FOUT + col0 + l] = acc[v];
  }
#else
  // Scalar fallback (same output mapping) — only if the builtin is missing.
  for (int v = 0; v < 8; ++v) {
    int r = row0 + v + half * 8;
    float sum = 0.f;
    for (int k = 0; k < FIN; ++k) sum += X[(long)r * FIN + k] * W[(long)k * FOUT + col0 + l];
    Hout[(long)r * FOUT + col0 + l] = sum;
  }
#endif
}

// al_s[n,h] = sum_c H[n,h,c]*a_s[h,c] ; likewise al_d
__global__ void gat_alpha(const float* __restrict__ H,
                          const float* __restrict__ a_s, const float* __restrict__ a_d,
                          float* __restrict__ al_s, float* __restrict__ al_d,
                          int Hh, int C) {
  int i = blockIdx.x * blockDim.x + threadIdx.x;    // n*Hh + h
  if (i >= N_NODES * Hh) return;
  int n = i / Hh, h = i % Hh;
  const float* hv = H + (long)n * Hh * C + h * C;
  float s = 0.f, d = 0.f;
  for (int c = 0; c < C; ++c) { float v = hv[c]; s += v * a_s[h * C + c]; d += v * a_d[h * C + c]; }
  al_s[i] = s; al_d[i] = d;
}

__global__ void gat_init(unsigned* __restrict__ m, float* __restrict__ s,
                         float* __restrict__ acc, int nm, int nacc) {
  int i = blockIdx.x * blockDim.x + threadIdx.x;
  if (i < nm) { m[i] = NEG_INF_KEY; s[i] = 0.f; }
  if (i < nacc) acc[i] = 0.f;
}

__global__ void gat_edge_max(const int* __restrict__ ei,
                             const float* __restrict__ al_s, const float* __restrict__ al_d,
                             unsigned* __restrict__ m, int Hh) {
  int e = blockIdx.x * blockDim.x + threadIdx.x;
  if (e >= E_TOT) return;
  int s, d; edge_sd(ei, e, s, d);
  for (int h = 0; h < Hh; ++h) {
    float v = leaky02(al_s[s * Hh + h] + al_d[d * Hh + h]);
    atomicMax(&m[d * Hh + h], flipf(v));
  }
}

__global__ void gat_edge_sum(const int* __restrict__ ei,
                             const float* __restrict__ al_s, const float* __restrict__ al_d,
                             const unsigned* __restrict__ m, float* __restrict__ sden, int Hh) {
  int e = blockIdx.x * blockDim.x + threadIdx.x;
  if (e >= E_TOT) return;
  int s, d; edge_sd(ei, e, s, d);
  for (int h = 0; h < Hh; ++h) {
    float v  = leaky02(al_s[s * Hh + h] + al_d[d * Hh + h]);
    float mx = unflipu(m[d * Hh + h]);
    atomicAdd(&sden[d * Hh + h], expf(v - mx));
  }
}

// One thread per (edge, 4-channel group): float4 gather of h[src], 4 atomic adds
// into acc[dst]. acc fits in L2 (<=48MB) so atomics stay on-chip.
__global__ void gat_edge_aggr(const int* __restrict__ ei, const float* __restrict__ Hf,
                              const float* __restrict__ al_s, const float* __restrict__ al_d,
                              const unsigned* __restrict__ m, const float* __restrict__ sden,
                              float* __restrict__ acc, int Hh, int C) {
  const int groups = (Hh * C) / 4;
  long tid = (long)blockIdx.x * blockDim.x + threadIdx.x;
  long e = tid / groups;
  int  g = (int)(tid % groups);
  if (e >= E_TOT) return;
  int s, d; edge_sd(ei, (int)e, s, d);
  int ch = g * 4;
  int h  = ch / C;                                  // group never straddles heads (C%4==0)
  float v  = leaky02(al_s[s * Hh + h] + al_d[d * Hh + h]);
  float mx = unflipu(m[d * Hh + h]);
  float alpha = expf(v - mx) / sden[d * Hh + h];
  const float4 hv = *(const float4*)(Hf + (long)s * Hh * C + ch);
  float* out = acc + (long)d * Hh * C + ch;
  atomicAdd(out + 0, hv.x * alpha);
  atomicAdd(out + 1, hv.y * alpha);
  atomicAdd(out + 2, hv.z * alpha);
  atomicAdd(out + 3, hv.w * alpha);
}

// Layers 1/2: x_next = elu(acc + b)   (concat heads)
__global__ void gat_finalize_concat(const float* __restrict__ acc, const float* __restrict__ b,
                                    float* __restrict__ xnext, int HC) {
  int i = blockIdx.x * blockDim.x + threadIdx.x;
  if (i >= N_NODES * HC) return;
  xnext[i] = elu1(acc[i] + b[i % HC]);
}

// Layer 3: out = log_softmax(elu(mean_h(acc) + b3))
__global__ void gat_finalize_mean_lsm(const float* __restrict__ acc, const float* __restrict__ b,
                                      float* __restrict__ out, int Hh, int C) {
  int n = blockIdx.x * blockDim.x + threadIdx.x;
  if (n >= N_NODES) return;
  float t[40];
  float mx = -INFINITY;
  for (int c = 0; c < C; ++c) {
    float sum = 0.f;
    for (int h = 0; h < Hh; ++h) sum += acc[(long)n * Hh * C + h * C + c];
    float v = elu1(sum / (float)Hh + b[c]);
    t[c] = v;
    mx = fmaxf(mx, v);
  }
  float se = 0.f;
  for (int c = 0; c < C; ++c) se += expf(t[c] - mx);
  float lse = mx + logf(se);
  for (int c = 0; c < C; ++c) out[(long)n * C + c] = t[c] - lse;
}

// ---------------------------------------------------------------------------
extern "C" void kernel_launch(void* const* d_in, const int* in_sizes, int n_in,
                              void* d_out, int out_size, void* d_ws, size_t ws_size,
                              hipStream_t stream) {
  const float* x   = (const float*)d_in[0];
  const int*   ei  = (const int*)d_in[1];
  const float* W1  = (const float*)d_in[2];
  const float* a1s = (const float*)d_in[3];
  const float* a1d = (const float*)d_in[4];
  const float* b1  = (const float*)d_in[5];
  const float* W2  = (const float*)d_in[6];
  const float* a2s = (const float*)d_in[7];
  const float* a2d = (const float*)d_in[8];
  const float* b2  = (const float*)d_in[9];
  const float* W3  = (const float*)d_in[10];
  const float* a3s = (const float*)d_in[11];
  const float* a3d = (const float*)d_in[12];
  const float* b3  = (const float*)d_in[13];
  float* out = (float*)d_out;

  // Workspace layout (floats): ~114 MB total
  float*    ws    = (float*)d_ws;
  float*    Hbuf  = ws;                       // 50000*240 = 12,000,000
  float*    acc   = Hbuf + 12000000;          // 12,000,000
  float*    xbuf  = acc + 12000000;           //  3,200,000
  float*    als   = xbuf + 3200000;           //    300,000
  float*    ald   = als + 300000;             //    300,000
  unsigned* mbuf  = (unsigned*)(ald + 300000);//    300,000
  float*    sden  = (float*)(mbuf + 300000);  //    300,000

  const int TB = 256;
  const int edgeBlocks = (E_TOT + TB - 1) / TB;

  // ---------------- Layer 1: 128 -> 4x16, concat ----------------
  {
    const int Hh = 4, C = 16, FIN = 128, FOUT = 64;
    int nTiles = (N_NODES / 16) * (FOUT / 16);
    gat_gemm_wmma<FIN, FOUT><<<(nTiles + 3) / 4, 128, 0, stream>>>(x, W1, Hbuf, nTiles);
    gat_alpha<<<(N_NODES * Hh + TB - 1) / TB, TB, 0, stream>>>(Hbuf, a1s, a1d, als, ald, Hh, C);
    int nacc = N_NODES * Hh * C;
    gat_init<<<(nacc + TB - 1) / TB, TB, 0, stream>>>(mbuf, sden, acc, N_NODES * Hh, nacc);
    gat_edge_max<<<edgeBlocks, TB, 0, stream>>>(ei, als, ald, mbuf, Hh);
    gat_edge_sum<<<edgeBlocks, TB, 0, stream>>>(ei, als, ald, mbuf, sden, Hh);
    long aggrT = (long)E_TOT * ((Hh * C) / 4);
    gat_edge_aggr<<<(int)((aggrT + TB - 1) / TB), TB, 0, stream>>>(ei, Hbuf, als, ald, mbuf, sden, acc, Hh, C);
    gat_finalize_concat<<<(nacc + TB - 1) / TB, TB, 0, stream>>>(acc, b1, xbuf, Hh * C);
  }
  // ---------------- Layer 2: 64 -> 4x16, concat ----------------
  {
    const int Hh = 4, C = 16, FIN = 64, FOUT = 64;
    int nTiles = (N_NODES / 16) * (FOUT / 16);
    gat_gemm_wmma<FIN, FOUT><<<(nTiles + 3) / 4, 128, 0, stream>>>(xbuf, W2, Hbuf, nTiles);
    gat_alpha<<<(N_NODES * Hh + TB - 1) / TB, TB, 0, stream>>>(Hbuf, a2s, a2d, als, ald, Hh, C);
    int nacc = N_NODES * Hh * C;
    gat_init<<<(nacc + TB - 1) / TB, TB, 0, stream>>>(mbuf, sden, acc, N_NODES * Hh, nacc);
    gat_edge_max<<<edgeBlocks, TB, 0, stream>>>(ei, als, ald, mbuf, Hh);
    gat_edge_sum<<<edgeBlocks, TB, 0, stream>>>(ei, als, ald, mbuf, sden, Hh);
    long aggrT = (long)E_TOT * ((Hh * C) / 4);
    gat_edge_aggr<<<(int)((aggrT + TB - 1) / TB), TB, 0, stream>>>(ei, Hbuf, als, ald, mbuf, sden, acc, Hh, C);
    gat_finalize_concat<<<(nacc + TB - 1) / TB, TB, 0, stream>>>(acc, b2, xbuf, Hh * C);
  }
  // ---------------- Layer 3: 64 -> 6x40, mean + log_softmax ----------------
  {
    const int Hh = 6, C = 40, FIN = 64, FOUT = 240;
    int nTiles = (N_NODES / 16) * (FOUT / 16);
    gat_gemm_wmma<FIN, FOUT><<<(nTiles + 3) / 4, 128, 0, stream>>>(xbuf, W3, Hbuf, nTiles);
    gat_alpha<<<(N_NODES * Hh + TB - 1) / TB, TB, 0, stream>>>(Hbuf, a3s, a3d, als, ald, Hh, C);
    int nacc = N_NODES * Hh * C;
    gat_init<<<(nacc + TB - 1) / TB, TB, 0, stream>>>(mbuf, sden, acc, N_NODES * Hh, nacc);
    gat_edge_max<<<edgeBlocks, TB, 0, stream>>>(ei, als, ald, mbuf, Hh);
    gat_edge_sum<<<edgeBlocks, TB, 0, stream>>>(ei, als, ald, mbuf, sden, Hh);
    long aggrT = (long)E_TOT * ((Hh * C) / 4);
    gat_edge_aggr<<<(int)((aggrT + TB - 1) / TB), TB, 0, stream>>>(ei, Hbuf, als, ald, mbuf, sden, acc, Hh, C);
    gat_finalize_mean_lsm<<<(N_NODES + TB - 1) / TB, TB, 0, stream>>>(acc, b3, out, Hh, C);
  }
}